// ScatterAttention_80109730005700
// MI455X (gfx1250) — compile-verified
//
#include <hip/hip_runtime.h>
#include <hip/hip_bf16.h>

// ---------------------------------------------------------------------------
// ScatterAttention for MI455X (gfx1250, wave32)
//   keys_proj = attn_keys @ W.T + b        (f32 WMMA 16x16x4 GEMM)
//   probs[n]  = dot(scattered_values[n], keys_proj[idx[n]])
//   segment softmax over idx, then scatter-sum of score-weighted values
// ---------------------------------------------------------------------------

typedef __attribute__((ext_vector_type(2))) float v2f;
typedef __attribute__((ext_vector_type(8))) float v8f;

#define D 128

// Monotone float <-> uint key for atomicMax-based segment max (non-NaN safe).
__device__ __forceinline__ unsigned int orderKey(float f) {
  unsigned int b = __float_as_uint(f);
  return (b & 0x80000000u) ? ~b : (b | 0x80000000u);
}
__device__ __forceinline__ float fromOrderKey(unsigned int k) {
  unsigned int b = (k & 0x80000000u) ? (k & 0x7FFFFFFFu) : ~k;
  return __uint_as_float(b);
}

// ---------------------------------------------------------------------------
// Kernel 1: keys_proj[m, n] = sum_k attn_keys[m,k] * W[n,k] + b[n]
// One wave32 per 16x16 output tile, K looped in steps of 4 via
// V_WMMA_F32_16X16X4_F32.  A-frag: attn_keys rows; B-frag: W rows (== W.T cols).
// Per ISA layout, lane L holds (row = L&15) elements {k+2*(L>>4), +1} for both
// fragments -> a single float2 load each.
// ---------------------------------------------------------------------------
__global__ void keys_proj_wmma_kernel(const float* __restrict__ A,   // [M, 128]
                                      const float* __restrict__ W,   // [128, 128]
                                      const float* __restrict__ bias,// [128]
                                      float* __restrict__ keys_proj, // [M, 128]
                                      int M) {
  const int lane = threadIdx.x & 31;
  const int waveInBlock = threadIdx.x >> 5;
  const int wavesPerBlock = blockDim.x >> 5;
  const int tile = blockIdx.x * wavesPerBlock + waveInBlock;
  const int nTilesN = D / 16;                 // 8
  const int tm = tile / nTilesN;              // M-tile
  const int tn = tile % nTilesN;              // N-tile (output column block)
  if (tm * 16 >= M) return;                   // wave-uniform; EXEC stays all-1s

  const int l16 = lane & 15;
  const int hi  = lane >> 4;                  // 0: K,K+1   1: K+2,K+3

  const float* arow = A + (size_t)(tm * 16 + l16) * D + 2 * hi;
  const float* brow = W + (size_t)(tn * 16 + l16) * D + 2 * hi;

  v8f acc = {};
#pragma unroll
  for (int k = 0; k < D; k += 4) {
    v2f af = *(const v2f*)(arow + k);
    v2f bf = *(const v2f*)(brow + k);
    // 8 args: (neg_a, A, neg_b, B, c_mod, C, reuse_a, reuse_b)
    acc = __builtin_amdgcn_wmma_f32_16x16x4_f32(
        false, af, false, bf, (short)0, acc, false, false);
  }

  const float bv = bias[tn * 16 + l16];       // column value; same per half-lane
  // C/D layout: VGPR g -> row (g + 8*hi), lane l16 -> col l16
  float* out = keys_proj + (size_t)(tm * 16 + 8 * hi) * D + tn * 16 + l16;
#pragma unroll
  for (int g = 0; g < 8; ++g) out[(size_t)g * D] = acc[g] + bv;
}

// ---------------------------------------------------------------------------
// Kernel 2: per-row dot product + segment max (one wave32 per row)
// ---------------------------------------------------------------------------
__global__ void probs_kernel(const float* __restrict__ sv,        // [N, 128]
                             const int* __restrict__ idx,         // [N]
                             const float* __restrict__ keys_proj, // [M, 128]
                             float* __restrict__ probs,           // [N]
                             unsigned int* __restrict__ seg_max,  // [M]
                             int N) {
  const int row  = (int)((blockIdx.x * blockDim.x + threadIdx.x) >> 5);
  const int lane = threadIdx.x & 31;
  if (row >= N) return;

  const int seg = idx[row];
  const float4 v = ((const float4*)(sv + (size_t)row * D))[lane];
  const float4 k = ((const float4*)(keys_proj + (size_t)seg * D))[lane];
  float p = v.x * k.x + v.y * k.y + v.z * k.z + v.w * k.w;
#pragma unroll
  for (int off = 16; off > 0; off >>= 1) p += __shfl_xor(p, off, 32);

  if (lane == 0) {
    probs[row] = p;
    atomicMax(seg_max + seg, orderKey(p));
  }
}

// ---------------------------------------------------------------------------
// Kernel 3: ex = exp(p - segmax), in place; segment sum of ex
// ---------------------------------------------------------------------------
__global__ void exp_sum_kernel(float* __restrict__ probs,              // [N] in/out
                               const int* __restrict__ idx,            // [N]
                               const unsigned int* __restrict__ seg_max,// [M]
                               float* __restrict__ seg_sum,            // [M]
                               int N) {
  const int n = blockIdx.x * blockDim.x + threadIdx.x;
  if (n >= N) return;
  const int seg = idx[n];
  const float m = fromOrderKey(seg_max[seg]);
  const float e = expf(probs[n] - m);
  probs[n] = e;
  atomicAdd(seg_sum + seg, e);
}

// ---------------------------------------------------------------------------
// Kernel 4: scores + scatter-sum of score-weighted values (one wave per row)
// ---------------------------------------------------------------------------
__global__ void apply_kernel(const float* __restrict__ sv,       // [N, 128]
                             const int* __restrict__ idx,        // [N]
                             const float* __restrict__ ex,       // [N]
                             const float* __restrict__ seg_sum,  // [M]
                             float* __restrict__ scores,         // [N]
                             float* __restrict__ attn,           // [M, 128]
                             int N) {
  const int row  = (int)((blockIdx.x * blockDim.x + threadIdx.x) >> 5);
  const int lane = threadIdx.x & 31;
  if (row >= N) return;

  const int seg = idx[row];
  float s = 0.0f;
  if (lane == 0) s = ex[row] / seg_sum[seg];
  s = __shfl(s, 0, 32);
  if (lane == 0) scores[row] = s;

  const float4 v = ((const float4*)(sv + (size_t)row * D))[lane];
  float* dst = attn + (size_t)seg * D + lane * 4;
  atomicAdd(dst + 0, v.x * s);
  atomicAdd(dst + 1, v.y * s);
  atomicAdd(dst + 2, v.z * s);
  atomicAdd(dst + 3, v.w * s);
}

// ---------------------------------------------------------------------------
// Launcher
//   d_in[0] scattered_values f32 [N,128]
//   d_in[1] indices          i32 [N]      (JAX demotes int64 -> int32)
//   d_in[2] attn_keys        f32 [M,128]
//   d_in[3] W                f32 [128,128]
//   d_in[4] b                f32 [128]
//   d_out = [scores (N floats)] ++ [attn_applied (M*128 floats)]
// ---------------------------------------------------------------------------
extern "C" void kernel_launch(void* const* d_in, const int* in_sizes, int n_in,
                              void* d_out, int out_size, void* d_ws, size_t ws_size,
                              hipStream_t stream) {
  const float* sv   = (const float*)d_in[0];
  const int*   idx  = (const int*)d_in[1];
  const float* ak   = (const float*)d_in[2];
  const float* W    = (const float*)d_in[3];
  const float* bias = (const float*)d_in[4];

  const int N = in_sizes[0] / D;   // 1,000,000
  const int M = in_sizes[2] / D;   // 50,000

  float* scores = (float*)d_out;
  float* attn   = (float*)d_out + (size_t)N;

  // Workspace layout
  float*        keys_proj = (float*)d_ws;                      // M*128 f32
  float*        probs     = keys_proj + (size_t)M * D;         // N f32 (then ex)
  unsigned int* seg_max   = (unsigned int*)(probs + (size_t)N);// M u32
  float*        seg_sum   = (float*)(seg_max + (size_t)M);     // M f32

  // Init: seg_max = 0 (ordered-key floor), seg_sum = 0, attn output = 0.
  hipMemsetAsync(seg_max, 0, (size_t)M * sizeof(unsigned int), stream);
  hipMemsetAsync(seg_sum, 0, (size_t)M * sizeof(float), stream);
  hipMemsetAsync(attn,    0, (size_t)M * D * sizeof(float), stream);

  // 1) WMMA GEMM: keys_proj = attn_keys @ W.T + b
  {
    const int tiles = ((M + 15) / 16) * (D / 16);   // 25,000
    const int wavesPerBlock = 8;                    // 256 threads
    const int blocks = (tiles + wavesPerBlock - 1) / wavesPerBlock;
    keys_proj_wmma_kernel<<<blocks, wavesPerBlock * 32, 0, stream>>>(
        ak, W, bias, keys_proj, M);
  }

  // 2) probs + segment max (one wave per row)
  {
    const int rowsPerBlock = 8;                     // 256 threads = 8 waves
    const int blocks = (N + rowsPerBlock - 1) / rowsPerBlock;
    probs_kernel<<<blocks, 256, 0, stream>>>(sv, idx, keys_proj, probs, seg_max, N);
  }

  // 3) exp + segment sum
  exp_sum_kernel<<<(N + 255) / 256, 256, 0, stream>>>(probs, idx, seg_max, seg_sum, N);

  // 4) scores + weighted scatter-sum
  {
    const int rowsPerBlock = 8;
    const int blocks = (N + rowsPerBlock - 1) / rowsPerBlock;
    apply_kernel<<<blocks, 256, 0, stream>>>(sv, idx, probs, seg_sum, scores, attn, N);
  }
}